// DeBut_70153995812927
// MI455X (gfx1250) — compile-verified
//
#include <hip/hip_runtime.h>
#include <hip/hip_bf16.h>

typedef float v2f __attribute__((ext_vector_type(2)));
typedef float v8f __attribute__((ext_vector_type(8)));

// ---------------------------------------------------------------------------
// Stage-pair fusion: factors (2s, 2s+1) of the butterfly chain combine into a
// dense 16x16 matrix per group.  p = 2s, d1 = 4^p, d2 = 4^(p+1).
//   group g -> h = g/d1, l = g%d1; element e in [0,16) sits at
//   n(e) = h*16*d1 + e*d1 + l   within the 4096-wide row.
//   W[j=(r1,r0)][k=(c1,c0)] = t2[h, r1, c1, r0*d1 + l] * t1[4h + c1, r0, c0, l]
// with t[z,r,c,dd] = tw[base + ((z*d + dd)*4 + r)*4 + c].
// Stored in V_WMMA_F32_16X16X4_F32 B-fragment layout (B = W^T), 8 floats/lane:
//   flat = ((s*256 + g)*32 + lane)*8 + kb*2 + reg
//   lane = j + 16*((k%4)/2), reg = (k%4)%2, kb = k/4.
// ---------------------------------------------------------------------------
__global__ __launch_bounds__(256) void debut_precompute_W(const float* __restrict__ tw,
                                                          float* __restrict__ wsW) {
    int tid = blockIdx.x * 256 + threadIdx.x;      // 3 * 256 * 256 = 196608 entries
    int s   = tid >> 16;                           // 0..2
    int rem = tid & 65535;
    int g   = rem >> 8;                            // 0..255
    int jk  = rem & 255;
    int j   = jk >> 4;                             // output index (r1,r0)
    int k   = jk & 15;                             // input index  (c1,c0)

    int d1 = 1 << (4 * s);                         // 1, 16, 256
    int d2 = d1 << 2;
    int h  = g / d1;
    int l  = g - h * d1;

    int r1 = j >> 2, r0 = j & 3;
    int c1 = k >> 2, c0 = k & 3;

    int base1 = (2 * s) * 16384;
    int base2 = base1 + 16384;

    float m1 = tw[base1 + (((4 * h + c1) * d1 + l) * 4 + r0) * 4 + c0];
    float m2 = tw[base2 + ((h * d2 + (r0 * d1 + l)) * 4 + r1) * 4 + c1];
    float w  = m2 * m1;

    int kb   = k >> 2;
    int kp   = k & 3;
    int lane = j + ((kp >> 1) << 4);
    int reg  = kp & 1;
    wsW[((size_t)((s * 256 + g) * 32 + lane)) * 8 + kb * 2 + reg] = w;
}

// ---------------------------------------------------------------------------
// Main fused kernel: 16 batch rows per block staged in LDS (padded stride to
// avoid bank conflicts), 3 in-place 16x16 butterfly stages via chained
// V_WMMA_F32_16X16X4_F32, then bias add + store.  512 threads = 16 wave32s;
// each wave owns 16 disjoint groups per stage -> no intra-stage barriers.
// ---------------------------------------------------------------------------
#define ROWS 16
#define MS   4100   // LDS row stride in floats (4096 + 4 pad)

__global__ __launch_bounds__(512) void debut_main(const float* __restrict__ x,
                                                  const float* __restrict__ wsW,
                                                  const float* __restrict__ bias,
                                                  float* __restrict__ out) {
    extern __shared__ float lds[];

    const int rbase = blockIdx.x * ROWS;
    const int tid   = threadIdx.x;

    // ---- Load 16 rows x 4096 f32, float4-coalesced ----
#pragma unroll
    for (int i = 0; i < 32; ++i) {
        int f    = tid + i * 512;          // float4 index in tile [0, 16384)
        int row  = f >> 10;                // 1024 float4 per row
        int col4 = f & 1023;
        float4 v = reinterpret_cast<const float4*>(x + (size_t)(rbase + row) * 4096)[col4];
        *reinterpret_cast<float4*>(&lds[row * MS + col4 * 4]) = v;
    }
    __syncthreads();

    const int wave = tid >> 5;
    const int lane = tid & 31;
    const int m_a  = lane & 15;            // A-matrix row (batch row within tile)
    const int kk0  = (lane >> 4) << 1;     // A-matrix K pair: lanes<16 -> K{0,1}, else K{2,3}

#pragma unroll
    for (int s = 0; s < 3; ++s) {
        const int d1 = 1 << (4 * s);       // element stride within a group: 1, 16, 256

        for (int gi = 0; gi < 16; ++gi) {
            const int g = wave * 16 + gi;  // 16 waves x 16 = 256 groups
            const int h = g / d1;
            const int l = g - h * d1;
            const int nbase = h * 16 * d1 + l;

            // ---- A fragments: 4 K-blocks, 2 regs each (ISA 16x4 f32 A layout) ----
            float a[8];
#pragma unroll
            for (int kb = 0; kb < 4; ++kb) {
                const int k0 = 4 * kb + kk0;
                a[2 * kb]     = lds[m_a * MS + nbase + k0 * d1];
                a[2 * kb + 1] = lds[m_a * MS + nbase + (k0 + 1) * d1];
            }

            // ---- B fragments: 8 consecutive floats per lane from L2-resident W ----
            const float4* bp = reinterpret_cast<const float4*>(
                wsW + ((size_t)((s * 256 + g) * 32 + lane)) * 8);
            float4 b01 = bp[0];
            float4 b23 = bp[1];

            // ---- D = A x B (+C), chained K=4 WMMAs for K=16, full f32 ----
            v8f acc = {};
            {
                v2f a0 = {a[0], a[1]}, b0 = {b01.x, b01.y};
                acc = __builtin_amdgcn_wmma_f32_16x16x4_f32(false, a0, false, b0,
                                                            (short)0, acc, false, false);
            }
            {
                v2f a1 = {a[2], a[3]}, b1 = {b01.z, b01.w};
                acc = __builtin_amdgcn_wmma_f32_16x16x4_f32(false, a1, false, b1,
                                                            (short)0, acc, false, false);
            }
            {
                v2f a2 = {a[4], a[5]}, b2 = {b23.x, b23.y};
                acc = __builtin_amdgcn_wmma_f32_16x16x4_f32(false, a2, false, b2,
                                                            (short)0, acc, false, false);
            }
            {
                v2f a3 = {a[6], a[7]}, b3 = {b23.z, b23.w};
                acc = __builtin_amdgcn_wmma_f32_16x16x4_f32(false, a3, false, b3,
                                                            (short)0, acc, false, false);
            }

            // ---- Write D back in place (ISA 16x16 f32 C/D layout) ----
            const int jcol = lane & 15;
            const int mhi  = (lane >> 4) << 3;
#pragma unroll
            for (int r = 0; r < 8; ++r) {
                lds[(mhi + r) * MS + nbase + jcol * d1] = acc[r];
            }
        }
        __syncthreads();
    }

    // ---- Bias add + store, float4-coalesced ----
#pragma unroll
    for (int i = 0; i < 32; ++i) {
        int f    = tid + i * 512;
        int row  = f >> 10;
        int col4 = f & 1023;
        float4 v  = *reinterpret_cast<const float4*>(&lds[row * MS + col4 * 4]);
        float4 bb = reinterpret_cast<const float4*>(bias)[col4];
        v.x += bb.x; v.y += bb.y; v.z += bb.z; v.w += bb.w;
        reinterpret_cast<float4*>(out + (size_t)(rbase + row) * 4096)[col4] = v;
    }
}

extern "C" void kernel_launch(void* const* d_in, const int* in_sizes, int n_in,
                              void* d_out, int out_size, void* d_ws, size_t ws_size,
                              hipStream_t stream) {
    const float* x       = (const float*)d_in[0];   // (8192, 4096) f32
    const float* twiddle = (const float*)d_in[1];   // (98304,)     f32
    const float* bias    = (const float*)d_in[2];   // (4096,)      f32
    float*       out     = (float*)d_out;           // (8192, 4096) f32
    float*       wsW     = (float*)d_ws;            // 3*256*256 f32 = 768 KB

    // Build fused 16x16 stage matrices in WMMA B-fragment layout.
    debut_precompute_W<<<768, 256, 0, stream>>>(twiddle, wsW);

    // Fully fused butterfly: 1 read + 1 write of x per element.
    const size_t lds_bytes = (size_t)ROWS * MS * sizeof(float);  // 262,400 B
    debut_main<<<8192 / ROWS, 512, lds_bytes, stream>>>(x, wsW, bias, out);
}